// EvoAttentionCausalTorch_2190433321005
// MI455X (gfx1250) — compile-verified
//
#include <hip/hip_runtime.h>
#include <hip/hip_bf16.h>
#include <math.h>

typedef __attribute__((ext_vector_type(2))) float v2f;
typedef __attribute__((ext_vector_type(8))) float v8f;

#define S_LEN 4096
#define D_DIM 64
#define NCHUNK 16
#define CHUNK 256   /* S_LEN / NCHUNK */
#define BH 64       /* B*H */

__device__ __forceinline__ float sigmoidf_fast(float x) {
    // v_rcp_f32 + v_exp_f32: avoids the IEEE div_fmas/div_fixup expansion
    return __builtin_amdgcn_rcpf(1.0f + __expf(-x));
}

// ---------------- Pass 1: per-(bh,chunk) channel sums of v_swiglu ----------------
// float4 streaming: each iteration the block covers 8 rows x 64 channels.
__global__ void pass1_chunk_sums(const float* __restrict__ V, float* __restrict__ ws) {
    int bc    = blockIdx.x;        // 0..BH*NCHUNK-1
    int bh    = bc / NCHUNK;
    int chunk = bc % NCHUNK;
    int t     = threadIdx.x;       // 0..127
    int rg    = t >> 4;            // row-in-group 0..7
    int cg    = t & 15;            // 4-channel column group
    const float* vb = V + ((size_t)bh * S_LEN + (size_t)chunk * CHUNK) * D_DIM;

    float4 acc = make_float4(0.f, 0.f, 0.f, 0.f);
    for (int i = 0; i < CHUNK / 8; ++i) {
        const float* p = vb + (size_t)(i * 8 + rg) * D_DIM + cg * 4;
        __builtin_prefetch(p + 16 * D_DIM, 0, 3);           // global_prefetch_b8
        float4 v4 = *(const float4*)p;
        acc.x += v4.x * v4.x * sigmoidf_fast(v4.x);
        acc.y += v4.y * v4.y * sigmoidf_fast(v4.y);
        acc.z += v4.z * v4.z * sigmoidf_fast(v4.z);
        acc.w += v4.w * v4.w * sigmoidf_fast(v4.w);
    }
    __shared__ __align__(16) float part[128 * 4];
    *(float4*)&part[t * 4] = acc;
    __syncthreads();
    if (t < 64) {
        // channel t partials live at part[t + 64*j], j=0..7 (conflict-free)
        float s = 0.0f;
#pragma unroll
        for (int j = 0; j < 8; ++j) s += part[t + 64 * j];
        ws[(size_t)(bh * NCHUNK + chunk) * D_DIM + t] = s;
    }
}

// ---------------- Pass 2: in-place exclusive scan over chunks per (bh,d) ----------------
__global__ void pass2_scan(float* __restrict__ ws) {
    int bh = blockIdx.x;    // 0..63
    int d  = threadIdx.x;   // 0..63
    float run = 0.0f;
    for (int c = 0; c < NCHUNK; ++c) {
        size_t idx = (size_t)(bh * NCHUNK + c) * D_DIM + d;
        float tmp = ws[idx];
        ws[idx]   = run;
        run      += tmp;
    }
}

// ---------------- Pass 3: WMMA tile-cumsum + norms/gating/output ----------------
// 128 threads = 4 waves. Wave w owns channels [16w, 16w+16). Cumsum within each
// 16-row tile is L16 (lower-triangular ones) @ X via v_wmma_f32_16x16x4_f32,
// with the running per-channel carry preloaded into C; carry hand-off between
// tiles via ds_bpermute.
__global__ void pass3_main(const float* __restrict__ Q, const float* __restrict__ K,
                           const float* __restrict__ V, const float* __restrict__ prefix,
                           float* __restrict__ out) {
    int bc    = blockIdx.x;
    int bh    = bc / NCHUNK;
    int chunk = bc % NCHUNK;
    int tid   = threadIdx.x;      // 0..127
    int w     = tid >> 5;         // wave id = channel tile
    int L     = tid & 31;         // lane
    int lh    = L >> 4;           // lane half (K pair for A/B, M half for C)
    int n     = L & 15;           // M for A, N for B/C
    int kbase = lh * 2;

    __shared__ __align__(16) float Vlds[16 * 64];
    __shared__ __align__(16) float Clds[16 * 64];
    __shared__ float red[3 * 128];

    // A operand: L16[m][k] = (k <= m), four 16x4 K-slices.
    // 32-bit A layout: lanes 0-15 hold K=0,1 (vgpr0,1); lanes 16-31 hold K=2,3.
    v2f A[4];
#pragma unroll
    for (int kc = 0; kc < 4; ++kc)
#pragma unroll
        for (int j = 0; j < 2; ++j)
            A[kc][j] = ((4 * kc + kbase + j) <= n) ? 1.0f : 0.0f;

    float carry = prefix[(size_t)(bh * NCHUNK + chunk) * D_DIM + w * 16 + n];

    size_t chunk_base = ((size_t)bh * S_LEN + (size_t)chunk * CHUNK) * D_DIM;
    int erow = tid >> 3;          // epilogue: 8 threads per row
    int ecb  = (tid & 7) * 8;     // 8 channels each

    for (int tile = 0; tile < CHUNK / 16; ++tile) {
        size_t tile_base = chunk_base + (size_t)tile * 16 * D_DIM;

        // Prefetch next tile (V for the matmul, Q/K for the epilogue)
        if (tile + 1 < CHUNK / 16) {
            size_t nb = tile_base + 16 * D_DIM + (size_t)tid * 8;
            __builtin_prefetch(V + nb, 0, 3);
            __builtin_prefetch(Q + nb, 0, 3);
            __builtin_prefetch(K + nb, 0, 3);
        }

        // Cooperative 16x64 V tile load -> LDS (2 x b128 per thread)
        {
            const float4* src = (const float4*)(V + tile_base) + tid * 2;
            float4 a0 = src[0];
            float4 a1 = src[1];
            float4* dst = (float4*)Vlds + tid * 2;
            dst[0] = a0;
            dst[1] = a1;
        }
        __syncthreads();

        // B operand from LDS (= silu(v)*v), layout mirrors A's K split.
        v2f Bop[4];
#pragma unroll
        for (int kc = 0; kc < 4; ++kc)
#pragma unroll
            for (int j = 0; j < 2; ++j) {
                int row = 4 * kc + kbase + j;
                float v = Vlds[row * 64 + w * 16 + n];
                Bop[kc][j] = v * v * sigmoidf_fast(v);
            }

        // C init: carry broadcast down this lane's column.
        v8f acc;
#pragma unroll
        for (int r = 0; r < 8; ++r) acc[r] = carry;

        // cumsum-as-matmul: acc = L16 @ X + carry  (4 fp32 WMMA K-steps)
#pragma unroll
        for (int kc = 0; kc < 4; ++kc)
            acc = __builtin_amdgcn_wmma_f32_16x16x4_f32(
                false, A[kc], false, Bop[kc], (short)0, acc, false, false);

        // New carry = row 15 (vgpr7, lanes 16..31) broadcast to same-column lanes.
        carry = __int_as_float(
            __builtin_amdgcn_ds_bpermute((16 + n) * 4, __float_as_int(acc[7])));

        // Spill C tile (lane L vgpr r holds element (m = r + 8*lh, col = 16w+n))
#pragma unroll
        for (int r = 0; r < 8; ++r)
            Clds[(r + 8 * lh) * 64 + w * 16 + n] = acc[r];
        __syncthreads();

        // ---- Epilogue: 8 threads per row, 8 channels each (all b128) ----
        size_t gbase = tile_base + (size_t)erow * D_DIM + ecb;

        float4 c0 = *(const float4*)&Clds[erow * 64 + ecb];
        float4 c1 = *(const float4*)&Clds[erow * 64 + ecb + 4];
        float4 v0 = *(const float4*)&Vlds[erow * 64 + ecb];
        float4 v1 = *(const float4*)&Vlds[erow * 64 + ecb + 4];
        float4 q0 = *(const float4*)(Q + gbase);
        float4 q1 = *(const float4*)(Q + gbase + 4);
        float4 k0 = *(const float4*)(K + gbase);
        float4 k1 = *(const float4*)(K + gbase + 4);

        float cv[8] = {c0.x, c0.y, c0.z, c0.w, c1.x, c1.y, c1.z, c1.w};
        float vv[8] = {v0.x, v0.y, v0.z, v0.w, v1.x, v1.y, v1.z, v1.w};
        float qv[8] = {q0.x, q0.y, q0.z, q0.w, q1.x, q1.y, q1.z, q1.w};
        float kv[8] = {k0.x, k0.y, k0.z, k0.w, k1.x, k1.y, k1.z, k1.w};

        float sc2 = 0.f, sq2 = 0.f, sk2 = 0.f;
#pragma unroll
        for (int j = 0; j < 8; ++j) {
            sc2 += cv[j] * cv[j];
            sq2 += qv[j] * qv[j];
            sk2 += kv[j] * kv[j];
        }
        red[tid]       = sc2;
        red[128 + tid] = sq2;
        red[256 + tid] = sk2;
        __syncthreads();

        int rb = erow * 8;
        float C2 = 0.f, Q2 = 0.f, K2 = 0.f;
#pragma unroll
        for (int i = 0; i < 8; ++i) {
            C2 += red[rb + i];
            Q2 += red[128 + rb + i];
            K2 += red[256 + rb + i];
        }
        float inv_c = rsqrtf(C2 * (1.0f / 64.0f) + 1e-5f);               // RMS(c)
        float rgate = (sqrtf(Q2) + 1e-8f) + (sqrtf(K2) + 1e-8f) + 1.0f;
        float scale = inv_c * __builtin_amdgcn_rcpf(rgate);              // hoisted

        float o0[8];
        float so2 = 0.f;
#pragma unroll
        for (int j = 0; j < 8; ++j) {
            float m = cv[j] * scale;                                     // mstate
            o0[j]   = vv[j] * (1.0f + m * sigmoidf_fast(m));             // V + silu(m)*V
            so2    += o0[j] * o0[j];
        }
        __syncthreads();           // red reuse
        red[tid] = so2;
        __syncthreads();
        float O2 = 0.f;
#pragma unroll
        for (int i = 0; i < 8; ++i) O2 += red[rb + i];
        float inv_o = rsqrtf(O2 * (1.0f / 64.0f) + 1e-5f);               // final RMS

        float4 r0 = make_float4(o0[0] * inv_o, o0[1] * inv_o, o0[2] * inv_o, o0[3] * inv_o);
        float4 r1 = make_float4(o0[4] * inv_o, o0[5] * inv_o, o0[6] * inv_o, o0[7] * inv_o);
        *(float4*)(out + gbase)     = r0;
        *(float4*)(out + gbase + 4) = r1;
        __syncthreads();           // protect LDS reuse next tile
    }
}

extern "C" void kernel_launch(void* const* d_in, const int* in_sizes, int n_in,
                              void* d_out, int out_size, void* d_ws, size_t ws_size,
                              hipStream_t stream) {
    (void)in_sizes; (void)n_in; (void)out_size; (void)ws_size;
    const float* Q = (const float*)d_in[0];
    const float* K = (const float*)d_in[1];
    const float* V = (const float*)d_in[2];
    float* out = (float*)d_out;
    float* ws  = (float*)d_ws;    // BH*NCHUNK*D_DIM floats = 256 KB

    pass1_chunk_sums<<<BH * NCHUNK, 128, 0, stream>>>(V, ws);
    pass2_scan<<<BH, 64, 0, stream>>>(ws);
    pass3_main<<<BH * NCHUNK, 128, 0, stream>>>(Q, K, V, ws, out);
}